// LengthRegulator_49563922596696
// MI455X (gfx1250) — compile-verified
//
#include <hip/hip_runtime.h>
#include <hip/hip_bf16.h>

typedef __attribute__((ext_vector_type(2))) float v2f;
typedef __attribute__((ext_vector_type(8))) float v8f;

#define NB   16
#define LL   256
#define DD   384            // Cin for conv1
#define FF   384            // Cout (and Cin for conv2) -- both convs are 384->384
#define TT   2048
#define RR   (NB * LL)      // 4096 rows
#define KTOT (3 * 384)      // im2row K = 1152
#define LNEPS 1e-5f

// ---------------------------------------------------------------------------
// Weight transpose: w (Cout=384, Cin=384, 3) torch layout ->
// wT[(p*384 + f)*2 + j] where K index kk = k*384 + d, p = kk>>1, j = kk&1.
// Makes the WMMA B-fragment a single contiguous lane-coalesced b64 load.
// ---------------------------------------------------------------------------
__global__ void wtrans_kernel(const float* __restrict__ w, float* __restrict__ wT) {
    const int idx = blockIdx.x * 256 + threadIdx.x;     // 0 .. 3*384*384-1
    const int j = idx & 1;
    const int t = idx >> 1;
    const int f = t % FF;
    const int p = t / FF;
    const int kk = (p << 1) | j;
    const int k = kk / 384;
    const int d = kk - k * 384;
    wT[idx] = w[(size_t)f * (384 * 3) + d * 3 + k];
}

// ---------------------------------------------------------------------------
// Conv1d (kernel=3, pad=1) as implicit-im2row GEMM with f32 WMMA.
// One wave computes a 16x64 tile (4 accumulators); A fragment reused 4x.
// Double-buffered K-chunks of 4 so loads overlap the WMMA chain.
// xin: (N, L, 384) row-major.  wT: pair-transposed weights.  y: (R, 384).
// ---------------------------------------------------------------------------
__global__ void conv_gemm_wmma(const float* __restrict__ xin,
                               const float* __restrict__ wT,
                               const float* __restrict__ bias,
                               float* __restrict__ yout) {
    const int lane = threadIdx.x & 31;
    const int wv   = threadIdx.x >> 5;
    const int gw   = blockIdx.x * 8 + wv;               // global wave id

    const int CT = FF >> 6;                             // 6 column-groups of 64
    const int rt = gw / CT;
    const int cg = gw - rt * CT;
    const int r0 = rt << 4;
    const int f0 = cg << 6;

    const int m  = lane & 15;                           // M row / N col for lane
    const int kb = (lane >> 4) << 1;                    // K-pair base in chunk of 4

    const int r = r0 + m;
    const int n = r >> 8;                               // L == 256
    const int l = r & (LL - 1);

    v8f c[4] = {};

    for (int k = 0; k < 3; ++k) {
        const int   srcl  = l + k - 1;
        const bool  valid = (unsigned)srcl < (unsigned)LL;
        const int   lrow  = valid ? srcl : l;           // clamped, always in-range
        const float* arow = xin + (size_t)(n * LL + lrow) * 384 + kb;
        // float offset for (k, d=0, lane): (p*384 + f)*2 with p = k*192 + kb/2
        const float* bp   = wT + (size_t)(k * 192 + (kb >> 1)) * 768 + (f0 + m) * 2;

        // prologue: chunk d = 0
        v2f a0 = *(const v2f*)(arow);
        a0.x = valid ? a0.x : 0.0f;
        a0.y = valid ? a0.y : 0.0f;
        v2f b0[4];
        #pragma unroll
        for (int j = 0; j < 4; ++j)
            b0[j] = *(const v2f*)(bp + 32 * j);

        for (int d = 0; d < 384; d += 4) {
            const int d2 = d + 4;
            v2f a1, b1[4];
            if (d2 < 384) {                             // prefetch next chunk
                a1 = *(const v2f*)(arow + d2);
                a1.x = valid ? a1.x : 0.0f;
                a1.y = valid ? a1.y : 0.0f;
                const float* bq = bp + (size_t)(d2 >> 1) * 768;
                #pragma unroll
                for (int j = 0; j < 4; ++j)
                    b1[j] = *(const v2f*)(bq + 32 * j);
            }
            #pragma unroll
            for (int j = 0; j < 4; ++j)
                c[j] = __builtin_amdgcn_wmma_f32_16x16x4_f32(
                           false, a0, false, b0[j], (short)0, c[j], false, false);
            if (d2 < 384) {
                a0 = a1;
                #pragma unroll
                for (int j = 0; j < 4; ++j) b0[j] = b1[j];
            }
        }
    }

    // C/D layout: lanes 0-15 -> (M=v, N=lane); lanes 16-31 -> (M=8+v, N=lane-16)
    const int crow = r0 + ((lane >> 4) << 3);
    #pragma unroll
    for (int j = 0; j < 4; ++j) {
        const int ccol = f0 + 16 * j + m;
        const float bv = bias[ccol];
        #pragma unroll
        for (int v = 0; v < 8; ++v)
            yout[(size_t)(crow + v) * FF + ccol] = c[j][v] + bv;
    }
}

// ---------------------------------------------------------------------------
// Row-wise LayerNorm (biased variance) + ReLU.  One 128-thread block per row.
// ---------------------------------------------------------------------------
__global__ void ln_relu_kernel(const float* __restrict__ y,
                               const float* __restrict__ g,
                               const float* __restrict__ b,
                               float* __restrict__ out, int C) {
    __shared__ float s1[128];
    __shared__ float s2[128];
    const int row = blockIdx.x;
    const int tid = threadIdx.x;
    const float* yr = y + (size_t)row * C;

    float lsum = 0.0f, lsq = 0.0f;
    for (int i = tid; i < C; i += 128) {
        float v = yr[i];
        lsum += v;
        lsq  += v * v;
    }
    s1[tid] = lsum; s2[tid] = lsq;
    __syncthreads();
    for (int off = 64; off > 0; off >>= 1) {
        if (tid < off) { s1[tid] += s1[tid + off]; s2[tid] += s2[tid + off]; }
        __syncthreads();
    }
    const float mean = s1[0] / (float)C;
    const float var  = s2[0] / (float)C - mean * mean;
    const float inv  = rsqrtf(var + LNEPS);
    for (int i = tid; i < C; i += 128) {
        float v = (yr[i] - mean) * inv * g[i] + b[i];
        out[(size_t)row * C + i] = v > 0.0f ? v : 0.0f;
    }
}

// ---------------------------------------------------------------------------
// Final projection h @ lw.T + lb, ReLU -> dur_pred.  One wave per row.
// ---------------------------------------------------------------------------
__global__ void linear_relu_kernel(const float* __restrict__ h,
                                   const float* __restrict__ lw,
                                   const float* __restrict__ lb,
                                   float* __restrict__ out, int C) {
    const int gw   = (blockIdx.x * blockDim.x + threadIdx.x) >> 5;
    const int lane = threadIdx.x & 31;
    const float* hr = h + (size_t)gw * C;
    float s = 0.0f;
    for (int i = lane; i < C; i += 32) s += hr[i] * lw[i];
    #pragma unroll
    for (int off = 16; off > 0; off >>= 1) s += __shfl_xor(s, off, 32);
    if (lane == 0) {
        float v = s + lb[0];
        out[gw] = v > 0.0f ? v : 0.0f;
    }
}

// ---------------------------------------------------------------------------
// Per-batch inclusive scan of durations -> ends.  One 256-thread block per n.
// ---------------------------------------------------------------------------
__global__ void cumsum_kernel(const int* __restrict__ dur, int* __restrict__ ends) {
    __shared__ int s[LL];
    const int n = blockIdx.x, tid = threadIdx.x;
    s[tid] = dur[n * LL + tid];
    __syncthreads();
    for (int off = 1; off < LL; off <<= 1) {
        int v = (tid >= off) ? s[tid - off] : 0;
        __syncthreads();
        s[tid] += v;
        __syncthreads();
    }
    ends[n * LL + tid] = s[tid];
}

// ---------------------------------------------------------------------------
// Length regulator: out[n,t,:] = x[n,l(t),:] with l(t) = first l, ends[l] > t;
// zero past total duration.  One wave per (n,t) row; float4 streaming writes.
// ---------------------------------------------------------------------------
__global__ void length_reg_kernel(const float* __restrict__ x,
                                  const int* __restrict__ ends,
                                  float* __restrict__ out) {
    const int gw   = (blockIdx.x * blockDim.x + threadIdx.x) >> 5;
    const int lane = threadIdx.x & 31;
    const int n = gw >> 11;             // T == 2048
    const int t = gw & (TT - 1);

    const int* e = ends + n * LL;
    float4* dst = (float4*)(out + ((size_t)n * TT + t) * DD);
    const int total = e[LL - 1];

    if (t >= total) {
        float4 z; z.x = z.y = z.z = z.w = 0.0f;
        dst[lane] = z; dst[lane + 32] = z; dst[lane + 64] = z;
        return;
    }
    int lo = 0, hi = LL - 1;
    while (lo < hi) {
        int mid = (lo + hi) >> 1;
        if (e[mid] > t) hi = mid; else lo = mid + 1;
    }
    const float4* src = (const float4*)(x + ((size_t)n * LL + lo) * DD);
    dst[lane]      = src[lane];
    dst[lane + 32] = src[lane + 32];
    dst[lane + 64] = src[lane + 64];
}

// ---------------------------------------------------------------------------
extern "C" void kernel_launch(void* const* d_in, const int* in_sizes, int n_in,
                              void* d_out, int out_size, void* d_ws, size_t ws_size,
                              hipStream_t stream) {
    const float* x      = (const float*)d_in[0];
    const int*   target = (const int*)  d_in[1];
    // d_in[2] = mel_max_length scalar (fixed T=2048)
    const float* cw1 = (const float*)d_in[3];
    const float* cb1 = (const float*)d_in[4];
    const float* g1  = (const float*)d_in[5];
    const float* b1  = (const float*)d_in[6];
    const float* cw2 = (const float*)d_in[7];
    const float* cb2 = (const float*)d_in[8];
    const float* g2  = (const float*)d_in[9];
    const float* b2  = (const float*)d_in[10];
    const float* lw  = (const float*)d_in[11];
    const float* lb  = (const float*)d_in[12];

    float* out      = (float*)d_out;                       // (N,T,D)
    float* dur_pred = out + (size_t)NB * TT * DD;          // (N,L)

    // workspace layout
    int*   ends = (int*)d_ws;                              // N*L ints (16 KB slot)
    float* bufA = (float*)((char*)d_ws + 16384);           // R*F floats
    float* bufB = bufA + (size_t)RR * FF;                  // R*F floats
    float* wTb  = bufB + (size_t)RR * FF;                  // 3*384*384 floats

    const int nWaves     = (RR / 16) * (FF / 64);          // 1536 (16x64 per wave)
    const int gemmBlocks = nWaves / 8;                     // 192 blocks, 8 waves
    const int wtBlocks   = (3 * 384 * 384) / 256;          // 1728

    // variance predictor chain (wT buffer reused between the two convs;
    // stream ordering provides the dependency)
    wtrans_kernel<<<wtBlocks, 256, 0, stream>>>(cw1, wTb);
    conv_gemm_wmma<<<gemmBlocks, 256, 0, stream>>>(x, wTb, cb1, bufA);
    ln_relu_kernel<<<RR, 128, 0, stream>>>(bufA, g1, b1, bufB, FF);
    wtrans_kernel<<<wtBlocks, 256, 0, stream>>>(cw2, wTb);
    conv_gemm_wmma<<<gemmBlocks, 256, 0, stream>>>(bufB, wTb, cb2, bufA);
    ln_relu_kernel<<<RR, 128, 0, stream>>>(bufA, g2, b2, bufB, FF);
    linear_relu_kernel<<<RR / 8, 256, 0, stream>>>(bufB, lw, lb, dur_pred, FF);

    // length regulation
    cumsum_kernel<<<NB, LL, 0, stream>>>(target, ends);
    length_reg_kernel<<<(NB * TT) / 8, 256, 0, stream>>>(x, ends, out);
}